// SAGEModel_44848048505635
// MI455X (gfx1250) — compile-verified
//
#include <hip/hip_runtime.h>
#include <hip/hip_bf16.h>

typedef __attribute__((ext_vector_type(2))) float v2f;
typedef __attribute__((ext_vector_type(8))) float v8f;

#define D_FEAT 128
#define HALF   64

// ---------------------------------------------------------------- zero (float4)
__global__ void k_zero4(float4* __restrict__ p, int n4) {
    int i = blockIdx.x * blockDim.x + threadIdx.x;
    if (i < n4) p[i] = float4{0.f, 0.f, 0.f, 0.f};
}

// ---------------------------------------------------------------- degree count
__global__ void k_degree(const int* __restrict__ row, float* __restrict__ deg, int E) {
    int e = blockIdx.x * blockDim.x + threadIdx.x;
    if (e < E) atomicAdd(&deg[row[e]], 1.0f);
}

__global__ void k_invdeg(float* __restrict__ deg, int n) {
    int i = blockIdx.x * blockDim.x + threadIdx.x;
    if (i < n) deg[i] = 1.0f / fmaxf(deg[i], 1.0f);
}

// ---------------------------------------------------------------- edge scatter: one wave32 per edge
__global__ void k_scatter(const int* __restrict__ row, const int* __restrict__ col,
                          const float* __restrict__ h, float* __restrict__ nbr_sum, int E) {
    int wid  = (blockIdx.x * blockDim.x + threadIdx.x) >> 5;
    int lane = threadIdx.x & 31;
    if (wid >= E) return;
    int r = row[wid], c = col[wid];
    float4 v = ((const float4*)(h + (size_t)c * D_FEAT))[lane];  // 32 lanes * 4 = 128
    float* dst = nbr_sum + (size_t)r * D_FEAT + lane * 4;
    atomicAdd(dst + 0, v.x);
    atomicAdd(dst + 1, v.y);
    atomicAdd(dst + 2, v.z);
    atomicAdd(dst + 3, v.w);
}

// ---------------------------------------------------------------- fused SAGE layer
// out[n, 0:64]   = relu(h        @ Wself + bias[0:64])
// out[n, 64:128] = relu(nbr_mean @ Wnbr  + bias[64:128]),  nbr_mean = nbr_sum * inv_deg
// 4 waves/block, each wave: 16 rows x (2 halves x 64 cols). fp32 WMMA 16x16x4.
__global__ void k_sage(const float* __restrict__ hin, const float* __restrict__ nbr_sum,
                       const float* __restrict__ inv_deg,
                       const float* __restrict__ Wself, const float* __restrict__ Wnbr,
                       const float* __restrict__ bias, float* __restrict__ hout, int n) {
    __shared__ float sW[2][D_FEAT * HALF];   // 2 * 32KB = 64KB LDS
    const int tid = threadIdx.x;
    for (int i = tid; i < D_FEAT * HALF; i += blockDim.x) {
        sW[0][i] = Wself[i];
        sW[1][i] = Wnbr[i];
    }
    __syncthreads();

    const int wave = tid >> 5;
    const int lane = tid & 31;
    const int row_tile = blockIdx.x * (blockDim.x >> 5) + wave;
    if (row_tile * 16 >= n) return;
    const int row_base = row_tile * 16;

    const int m    = lane & 15;          // row (A) / col (B) within tile
    const int koff = (lane >> 4) << 1;   // lanes 0-15 -> K=0,1 ; lanes 16-31 -> K=2,3
    const int row  = row_base + m;
    const float scale = inv_deg[row];

    const float* hrow = hin     + (size_t)row * D_FEAT;
    const float* srow = nbr_sum + (size_t)row * D_FEAT;

    v8f acc[2][4] = {};
    for (int k = 0; k < D_FEAT; k += 4) {
        v2f aS, aN;
        aS.x = hrow[k + koff];
        aS.y = hrow[k + koff + 1];
        aN.x = srow[k + koff]     * scale;
        aN.y = srow[k + koff + 1] * scale;
#pragma unroll
        for (int t = 0; t < 4; ++t) {
            const int ncol = t * 16 + m;
            v2f bS, bN;
            bS.x = sW[0][(k + koff) * HALF + ncol];
            bS.y = sW[0][(k + koff + 1) * HALF + ncol];
            bN.x = sW[1][(k + koff) * HALF + ncol];
            bN.y = sW[1][(k + koff + 1) * HALF + ncol];
            acc[0][t] = __builtin_amdgcn_wmma_f32_16x16x4_f32(false, aS, false, bS,
                                                              (short)0, acc[0][t], false, false);
            acc[1][t] = __builtin_amdgcn_wmma_f32_16x16x4_f32(false, aN, false, bN,
                                                              (short)0, acc[1][t], false, false);
        }
    }

    // C/D layout: lanes 0-15: VGPR v -> M=v ; lanes 16-31: VGPR v -> M=v+8 ; N = lane&15
    const int moff = (lane >> 4) * 8;
#pragma unroll
    for (int hf = 0; hf < 2; ++hf)
#pragma unroll
        for (int t = 0; t < 4; ++t)
#pragma unroll
            for (int v = 0; v < 8; ++v) {
                const int r    = row_base + v + moff;
                const int cidx = hf * HALF + t * 16 + m;
                float val = acc[hf][t][v] + bias[cidx];
                hout[(size_t)r * D_FEAT + cidx] = fmaxf(val, 0.0f);
            }
}

// ---------------------------------------------------------------- generic fp32 WMMA GEMM (MLP head)
// C[n, NC] = (relu?) (A[n,K] @ B[K,NC] + bias). Each wave: 16 rows x NT*16 cols of one column block.
template <int NT>
__global__ void k_gemm(const float* __restrict__ A, const float* __restrict__ B,
                       const float* __restrict__ bias, float* __restrict__ C,
                       int n, int K, int NC, int ldc, int relu) {
    extern __shared__ float sB[];  // K * NT*16
    const int CT   = NT * 16;
    const int col0 = blockIdx.y * CT;
    const int tid  = threadIdx.x;
    for (int i = tid; i < K * CT; i += blockDim.x) {
        const int k = i / CT, c = i % CT;
        const int gc = col0 + c;
        sB[i] = (gc < NC) ? B[(size_t)k * NC + gc] : 0.0f;
    }
    __syncthreads();

    const int wave = tid >> 5;
    const int lane = tid & 31;
    const int row_tile = blockIdx.x * (blockDim.x >> 5) + wave;
    if (row_tile * 16 >= n) return;
    const int row_base = row_tile * 16;

    const int m    = lane & 15;
    const int koff = (lane >> 4) << 1;
    const float* arow = A + (size_t)(row_base + m) * K;

    v8f acc[NT] = {};
    for (int k = 0; k < K; k += 4) {
        v2f a;
        a.x = arow[k + koff];
        a.y = arow[k + koff + 1];
#pragma unroll
        for (int t = 0; t < NT; ++t) {
            v2f b;
            b.x = sB[(k + koff) * CT + t * 16 + m];
            b.y = sB[(k + koff + 1) * CT + t * 16 + m];
            acc[t] = __builtin_amdgcn_wmma_f32_16x16x4_f32(false, a, false, b,
                                                           (short)0, acc[t], false, false);
        }
    }

    const int moff = (lane >> 4) * 8;
#pragma unroll
    for (int t = 0; t < NT; ++t)
#pragma unroll
        for (int v = 0; v < 8; ++v) {
            const int r    = row_base + v + moff;
            const int cidx = col0 + t * 16 + m;
            if (cidx < NC) {
                float val = acc[t][v] + bias[cidx];
                if (relu) val = fmaxf(val, 0.0f);
                C[(size_t)r * ldc + cidx] = val;
            }
        }
}

// ---------------------------------------------------------------- launch
extern "C" void kernel_launch(void* const* d_in, const int* in_sizes, int n_in,
                              void* d_out, int out_size, void* d_ws, size_t ws_size,
                              hipStream_t stream) {
    const float* x     = (const float*)d_in[0];
    const int*   eidx  = (const int*)d_in[1];
    const float* selfK = (const float*)d_in[2];   // [3,128,64]
    const float* nbrK  = (const float*)d_in[3];   // [3,128,64]
    const float* bias  = (const float*)d_in[4];   // [3,128]
    const float* w1    = (const float*)d_in[5];   // [128,256]
    const float* b1    = (const float*)d_in[6];
    const float* w2    = (const float*)d_in[7];   // [256,40]
    const float* b2    = (const float*)d_in[8];
    float* out = (float*)d_out;

    const int N = in_sizes[0] / D_FEAT;           // 50000
    const int E = in_sizes[1] / 2;                // 800000
    const int* rowp = eidx;                       // targets
    const int* colp = eidx + E;                   // sources

    // workspace layout (floats)
    float* w       = (float*)d_ws;
    float* inv_deg = w;                           // N
    float* nbr     = inv_deg + 50048;             // N*128
    float* hA      = nbr + (size_t)N * D_FEAT;
    float* hB      = hA  + (size_t)N * D_FEAT;
    float* hidden  = hB  + (size_t)N * D_FEAT;    // N*256

    const int ND4 = (N * D_FEAT) / 4;
    const int row_tiles = (N + 15) / 16;
    const int gemm_blocks = (row_tiles + 3) / 4;

    // degrees -> inv_deg
    k_zero4<<<(N / 4 + 255) / 256, 256, 0, stream>>>((float4*)inv_deg, N / 4);
    k_degree<<<(E + 255) / 256, 256, 0, stream>>>(rowp, inv_deg, E);
    k_invdeg<<<(N + 255) / 256, 256, 0, stream>>>(inv_deg, N);

    const float* hin = x;
    float* houts[3] = { hA, hB, hA };
    for (int i = 0; i < 3; ++i) {
        k_zero4<<<(ND4 + 255) / 256, 256, 0, stream>>>((float4*)nbr, ND4);
        long long sthreads = (long long)E * 32;
        k_scatter<<<(int)((sthreads + 255) / 256), 256, 0, stream>>>(rowp, colp, hin, nbr, E);
        k_sage<<<gemm_blocks, 128, 0, stream>>>(hin, nbr, inv_deg,
                                                selfK + (size_t)i * D_FEAT * HALF,
                                                nbrK  + (size_t)i * D_FEAT * HALF,
                                                bias  + (size_t)i * D_FEAT,
                                                houts[i], N);
        hin = houts[i];
    }

    // MLP head: [N,128] @ [128,256] relu -> [N,256] @ [256,40]
    k_gemm<4><<<dim3(gemm_blocks, 4), 128, 128 * 64 * sizeof(float), stream>>>(
        hA, w1, b1, hidden, N, 128, 256, 256, 1);
    k_gemm<3><<<dim3(gemm_blocks, 1), 128, 256 * 48 * sizeof(float), stream>>>(
        hidden, w2, b2, out, N, 256, 40, 40, 0);
}